// SEAFLOW3DP_77034533421268
// MI455X (gfx1250) — compile-verified
//
#include <hip/hip_runtime.h>
#include <stdint.h>

typedef uint32_t u32;
typedef unsigned int v4u __attribute__((ext_vector_type(4)));
typedef int       v8i __attribute__((ext_vector_type(8)));
typedef int       v4i __attribute__((ext_vector_type(4)));
typedef float     v2f __attribute__((ext_vector_type(2)));

#define TILE 16

// pack flow[N,2,H,W] + dz[N,1,H,W] -> fld[N,3,H,W]
__global__ __launch_bounds__(256) void pack_fields(const float* __restrict__ flow,
                                                   const float* __restrict__ dz,
                                                   float* __restrict__ out, int HW) {
  const int n = blockIdx.y;
  for (int p = blockIdx.x * 256 + threadIdx.x; p < 3 * HW; p += gridDim.x * 256) {
    int c = p / HW, q = p - c * HW;
    out[(size_t)(n * 3 + c) * HW + q] =
        (c < 2) ? flow[(size_t)(n * 2 + c) * HW + q] : dz[(size_t)n * HW + q];
  }
}

// One 2x convex-upsample stage: fld[N,3,H,W] + mask[N,36,H,W] -> out[N,3,2H,2W]
// NT: use non-temporal stores for the final (never re-read) output tensor.
template <bool NT>
__global__ __launch_bounds__(256) void upsample_stage(const float* __restrict__ fld,
                                                      const float* __restrict__ mask,
                                                      float* __restrict__ out,
                                                      int H, int W) {
  __shared__ float smask[36 * TILE * TILE];  // [ch][y][x] tile, filled by TDM
  __shared__ float sfld[3 * 18 * 18];        // 3-ch halo tile, flow pre-scaled x2

  const int n  = blockIdx.z;
  const int h0 = blockIdx.y * TILE;
  const int w0 = blockIdx.x * TILE;
  const int tx = threadIdx.x, ty = threadIdx.y;
  const int tid = ty * TILE + tx;

#if __has_builtin(__builtin_amdgcn_tensor_load_to_lds)
  if (ty < 2) {  // wave 0 only (wave-uniform branch); TDM ignores EXEC
    const uint64_t ga =
        (uint64_t)(uintptr_t)(mask + ((size_t)n * 36 * H * W + (size_t)h0 * W + w0));
    const u32 lds = (u32)(uintptr_t)&smask[0];  // flat LDS addr low 32b == LDS offset
    v4u g0;
    g0[0] = 1u;                            // count=1, user mode, gather off
    g0[1] = lds;                           // lds_addr (bytes)
    g0[2] = (u32)ga;                       // global_addr[31:0]
    g0[3] = (u32)(ga >> 32) | (2u << 30);  // global_addr[56:32] | type=2 ("image")
    const u32 td0 = (u32)W, td1 = (u32)H;
    const uint64_t s0 = (uint64_t)W;       // dim0 stride = row pitch (elems)
    const uint64_t s1 = (uint64_t)H * W;   // dim1 stride = plane pitch (elems)
    v8i g1;
    g1[0] = (int)(2u << 16);                                     // data_size = 4B
    g1[1] = (int)((td0 & 0xFFFFu) << 16);                        // tensor_dim0 lo16
    g1[2] = (int)(((td0 >> 16) & 0xFFFFu) | ((td1 & 0xFFFFu) << 16));
    g1[3] = (int)(((td1 >> 16) & 0xFFFFu) | ((u32)TILE << 16));  // tile_dim0 = 16
    g1[4] = (int)((u32)TILE | (36u << 16));                      // tile_dim1=16, tile_dim2=36
    g1[5] = (int)(u32)s0;                                        // dim0_stride[31:0]
    g1[6] = (int)(((u32)(s0 >> 32) & 0xFFFFu) | (((u32)s1 & 0xFFFFu) << 16));
    g1[7] = (int)((u32)(s1 >> 16));                              // dim1_stride[47:16]
    const uint64_t s2 = (uint64_t)36 * H * W;
    v4i g2;
    g2[0] = 36;                                // tensor_dim2 (channels)
    g2[1] = 0;                                 // tensor_dim3
    g2[2] = (int)(u32)s2;                      // dim2 stride lo (tile_dim3=0: unused)
    g2[3] = (int)((u32)(s2 >> 32) & 0xFFFFu);  // dim2 stride hi | tile_dim3=0
    v4i g3 = {0, 0, 0, 0};
#if defined(__clang_major__) && __clang_major__ >= 23
    v8i gz = {0, 0, 0, 0, 0, 0, 0, 0};
    __builtin_amdgcn_tensor_load_to_lds(g0, g1, g2, g3, gz, 0);
#else
    __builtin_amdgcn_tensor_load_to_lds(g0, g1, g2, g3, 0);
#endif
  }
#else
  // Fallback: plain cooperative staging of the mask tile.
  for (int e = tid; e < 36 * TILE * TILE; e += 256) {
    int ch = e >> 8, p = e & 255;
    int yy = p >> 4, xx = p & 15;
    smask[e] = mask[(size_t)(n * 36 + ch) * H * W + (size_t)(h0 + yy) * W + (w0 + xx)];
  }
#endif

  // Field halo tile (zero padding at the image border); flow channels pre-scaled by 2.
  for (int i = tid; i < 3 * 18 * 18; i += 256) {
    int c = i / 324, r = i - c * 324;
    int yy = r / 18, xx = r - yy * 18;
    int gy = h0 + yy - 1, gx = w0 + xx - 1;
    float v = 0.0f;
    if (gy >= 0 && gy < H && gx >= 0 && gx < W)
      v = fld[((size_t)n * 3 + c) * H * W + (size_t)gy * W + gx];
    sfld[i] = (c < 2) ? 2.0f * v : v;
  }

#if __has_builtin(__builtin_amdgcn_s_wait_tensorcnt)
  __builtin_amdgcn_s_wait_tensorcnt(0);
#endif
  __syncthreads();

  // Softmax over the 9 taps for each of the 4 subpixels (channel = k*4 + i*2 + j).
  float m[9][4];
#pragma unroll
  for (int k = 0; k < 9; ++k)
#pragma unroll
    for (int ij = 0; ij < 4; ++ij)
      m[k][ij] = smask[(k * 4 + ij) * (TILE * TILE) + tid];

  float inv[4];
#pragma unroll
  for (int ij = 0; ij < 4; ++ij) {
    float mx = m[0][ij];
#pragma unroll
    for (int k = 1; k < 9; ++k) mx = fmaxf(mx, m[k][ij]);
    float s = 0.0f;
#pragma unroll
    for (int k = 0; k < 9; ++k) {
      m[k][ij] = __expf(m[k][ij] - mx);
      s += m[k][ij];
    }
    inv[ij] = __frcp_rn(s);
  }

  const int oh = 2 * (h0 + ty), ow = 2 * (w0 + tx);
  const size_t W2 = 2 * (size_t)W, H2 = 2 * (size_t)H;
#pragma unroll
  for (int c = 0; c < 3; ++c) {
    float a0 = 0.f, a1 = 0.f, a2 = 0.f, a3 = 0.f;
#pragma unroll
    for (int kh = 0; kh < 3; ++kh)
#pragma unroll
      for (int kw = 0; kw < 3; ++kw) {
        float p = sfld[c * 324 + (ty + kh) * 18 + (tx + kw)];
        int k = kh * 3 + kw;
        a0 = fmaf(m[k][0], p, a0);
        a1 = fmaf(m[k][1], p, a1);
        a2 = fmaf(m[k][2], p, a2);
        a3 = fmaf(m[k][3], p, a3);
      }
    size_t base = (((size_t)n * 3 + c) * H2 + oh) * W2 + ow;
    v2f r0 = {a0 * inv[0], a1 * inv[1]};
    v2f r1 = {a2 * inv[2], a3 * inv[3]};
    if (NT) {
      __builtin_nontemporal_store(r0, (v2f*)(out + base));
      __builtin_nontemporal_store(r1, (v2f*)(out + base + W2));
    } else {
      *(v2f*)(out + base)      = r0;
      *(v2f*)(out + base + W2) = r1;
    }
  }
}

extern "C" void kernel_launch(void* const* d_in, const int* in_sizes, int n_in,
                              void* d_out, int out_size, void* d_ws, size_t ws_size,
                              hipStream_t stream) {
  (void)in_sizes; (void)n_in; (void)out_size; (void)ws_size;
  const float* flow16 = (const float*)d_in[0];  // [4,2,48,64]
  const float* dz16   = (const float*)d_in[1];  // [4,1,48,64]
  const float* m16    = (const float*)d_in[2];  // [4,36,48,64]
  const float* m8     = (const float*)d_in[3];  // [4,36,96,128]
  const float* m4     = (const float*)d_in[4];  // [4,36,192,256]
  const float* m2     = (const float*)d_in[5];  // [4,36,384,512]
  float* outp = (float*)d_out;                  // [4,3,768,1024]

  const int N = 4;
  // Ping-pong workspace: A holds the 48x64 pack + 192x256 stage, B holds 96x128 + 384x512.
  float* bufA = (float*)d_ws;                      // <= 4*3*192*256 = 589,824 floats
  float* bufB = bufA + (size_t)4 * 3 * 192 * 256;  // <= 4*3*384*512 = 2,359,296 floats

  dim3 blk(TILE, TILE);
  pack_fields<<<dim3(36, N), 256, 0, stream>>>(flow16, dz16, bufA, 48 * 64);
  upsample_stage<false><<<dim3( 4,  3, N), blk, 0, stream>>>(bufA, m16, bufB,  48,  64);
  upsample_stage<false><<<dim3( 8,  6, N), blk, 0, stream>>>(bufB, m8,  bufA,  96, 128);
  upsample_stage<false><<<dim3(16, 12, N), blk, 0, stream>>>(bufA, m4,  bufB, 192, 256);
  upsample_stage<true ><<<dim3(32, 24, N), blk, 0, stream>>>(bufB, m2,  outp, 384, 512);
}